// MoEDense_10411000726246
// MI455X (gfx1250) — compile-verified
//
#include <hip/hip_runtime.h>

typedef __attribute__((ext_vector_type(2))) float v2f;
typedef __attribute__((ext_vector_type(4))) float v4f;
typedef __attribute__((ext_vector_type(8))) float v8f;

#define B_N 8192
#define D_N 128
#define F_N 128
#define T_N 64
#define TILE_M 16
#define LDA_S 132   // 128 + 4 pad: rows land on distinct LDS banks for column reads

__device__ __forceinline__ int clamp_t(int t) {
    return t < 0 ? 0 : (t >= T_N ? T_N - 1 : t);
}

__global__ void moe_zero(int* __restrict__ counts, int* __restrict__ cursor) {
    int t = threadIdx.x;
    if (t < T_N) { counts[t] = 0; cursor[t] = 0; }
}

__global__ void moe_hist(const int* __restrict__ idx, int* __restrict__ counts, int n) {
    int b = blockIdx.x * blockDim.x + threadIdx.x;
    if (b < n) atomicAdd(&counts[clamp_t(idx[b])], 1);
}

__global__ void moe_scan(const int* __restrict__ counts, int* __restrict__ offsets,
                         int* __restrict__ cursor) {
    if (threadIdx.x == 0) {
        int acc = 0;
        for (int t = 0; t < T_N; ++t) {
            offsets[t] = acc;
            cursor[t]  = acc;
            acc += counts[t];
        }
    }
}

__global__ void moe_scatter(const int* __restrict__ idx, int* __restrict__ cursor,
                            int* __restrict__ perm, int n) {
    int b = blockIdx.x * blockDim.x + threadIdx.x;
    if (b < n) {
        int p = atomicAdd(&cursor[clamp_t(idx[b])], 1);
        perm[p] = b;
    }
}

// One block = one (expert, 16-row tile). 8 wave32s; wave w owns output columns
// [w*16, w*16+16). fp32 WMMA 16x16x4, K-loop over D=128 (32 WMMA per tile).
__global__ __launch_bounds__(256)
void moe_gemm(const float* __restrict__ X, const float* __restrict__ Kall,
              const float* __restrict__ bias, const int* __restrict__ perm,
              const int* __restrict__ offsets, const int* __restrict__ counts,
              float* __restrict__ Y) {
    const int e    = blockIdx.y;
    const int cnt  = counts[e];
    const int t16  = blockIdx.x * TILE_M;
    if (t16 >= cnt) return;
    const int start = offsets[e];

    __shared__ float lda[TILE_M * LDA_S];

    // Stage gathered A tile (16 rows x 128 fp32) into LDS; float4 granularity.
    // 16 * 32 = 512 float4 slots, 256 threads -> 2 iterations, coalesced reads.
    for (int i = threadIdx.x; i < TILE_M * (D_N / 4); i += 256) {
        const int r    = i >> 5;       // row 0..15
        const int c4   = i & 31;       // float4 column
        const int ridx = t16 + r;
        v4f val = {0.f, 0.f, 0.f, 0.f};
        if (ridx < cnt) {
            const int g = perm[start + ridx];
            val = *(const v4f*)(X + (size_t)g * D_N + c4 * 4);
        }
        *(v4f*)(&lda[r * LDA_S + c4 * 4]) = val;
    }
    __syncthreads();

    const int lane = threadIdx.x & 31;
    const int wave = threadIdx.x >> 5;     // N-tile 0..7
    const int half = lane >> 4;            // 0: lanes 0-15, 1: lanes 16-31
    const int l16  = lane & 15;
    const int n    = wave * 16 + l16;      // global output column
    const int k0   = 2 * half;             // ISA 16x4 f32 layout: v0=K0/K2, v1=K1/K3

    const float* __restrict__ Ke = Kall + (size_t)e * D_N * F_N;

    v8f c = {0.f, 0.f, 0.f, 0.f, 0.f, 0.f, 0.f, 0.f};

    #pragma unroll 8
    for (int kk = 0; kk < D_N; kk += 4) {
        v2f a, b;
        // A 16x4: lane half selects K pair; row = l16 (padded LDS stride)
        a.x = lda[l16 * LDA_S + kk + k0];
        a.y = lda[l16 * LDA_S + kk + k0 + 1];
        // B 4x16: row K striped across lanes within a VGPR, column = n
        b.x = Ke[(size_t)(kk + k0) * F_N + n];
        b.y = Ke[(size_t)(kk + k0 + 1) * F_N + n];
        // 8 args: (neg_a, A, neg_b, B, c_mod, C, reuse_a, reuse_b)
        c = __builtin_amdgcn_wmma_f32_16x16x4_f32(
                false, a, false, b, (short)0, c, false, false);
    }

    const float bn = bias[e * F_N + n];

    // C/D layout: VGPR r -> M = r + 8*half, N = l16 (col n). Masked scatter store.
    #pragma unroll
    for (int r = 0; r < 8; ++r) {
        const int m    = r + 8 * half;
        const int ridx = t16 + m;
        if (ridx < cnt) {
            const int g = perm[start + ridx];
            Y[(size_t)g * F_N + n] = c[r] + bn;
        }
    }
}

extern "C" void kernel_launch(void* const* d_in, const int* in_sizes, int n_in,
                              void* d_out, int out_size, void* d_ws, size_t ws_size,
                              hipStream_t stream) {
    const float* X    = (const float*)d_in[0];   // [B, D]
    const int*   idx  = (const int*)d_in[1];     // [B]
    const float* Kall = (const float*)d_in[2];   // [T, D, F]
    const float* bias = (const float*)d_in[3];   // [T, F]
    float*       Y    = (float*)d_out;           // [B, F]

    // Workspace layout: perm[B] | counts[T] | offsets[T] | cursor[T]  (~33 KB)
    int* perm    = (int*)d_ws;
    int* counts  = perm + B_N;
    int* offsets = counts + T_N;
    int* cursor  = offsets + T_N;

    moe_zero<<<1, 64, 0, stream>>>(counts, cursor);
    moe_hist<<<(B_N + 255) / 256, 256, 0, stream>>>(idx, counts, B_N);
    moe_scan<<<1, 32, 0, stream>>>(counts, offsets, cursor);
    moe_scatter<<<(B_N + 255) / 256, 256, 0, stream>>>(idx, cursor, perm, B_N);

    dim3 grid((B_N + TILE_M - 1) / TILE_M, T_N);   // 512 tiles x 64 experts, early-exit
    moe_gemm<<<grid, 256, 0, stream>>>(X, Kall, bias, perm, offsets, counts, Y);
}